// KuramotoSystemV2_55439437857256
// MI455X (gfx1250) — compile-verified
//
#include <hip/hip_runtime.h>
#include <hip/hip_bf16.h>

// ---------------------------------------------------------------------------
// Kuramoto system on MI455X (gfx1250, wave32)
//
//   Phase 1: coupling[b,n,m] = clip(exp(-acosh(max(dot, 1+1e-7))/tau'), 0, 1)
//            where dot = E[n,0]E[m,0] - sum_{k>=1} E[n,k]E[m,k]
//            -> fp32 GEMM via V_WMMA_F32_16X16X4_F32 (precision-critical:
//               acosh near 1 ~ sqrt(2(arg-1)); fp16 inputs would be garbage)
//   Phase 2: 16 steps, each: Sc = C*cos(theta), Ss = C*sin(theta)
//            dtheta = omega + (K/N)(sin(th_n)*Sc - cos(th_n)*Ss)
//            bandwidth-bound matvec; 128MB coupling fits in 192MB L2.
// ---------------------------------------------------------------------------

typedef __attribute__((ext_vector_type(2))) float v2f;
typedef __attribute__((ext_vector_type(8))) float v8f;

#define KN_B 8
#define KN_N 2048
#define KN_D 128
#define KN_NT (KN_N / 16)        /* 128 column/row tiles  */
#define KN_TJ 16                 /* col tiles per wave    */
#define KN_GJ (KN_NT / KN_TJ)    /* 8 col-tile groups     */

#define INV_TAU (1.0f / (1.0f + 1e-6f))
#define TWO_PI_F 6.28318530717958647692f
#define DT_F 0.1f
#define K_OVER_N (1.0f / 2048.0f)

// ---------------------------------------------------------------------------
// Kernel 1: build coupling matrix with fp32 WMMA.
// One wave (32 threads) owns a 16-row tile and sweeps 16 column tiles,
// caching the sign-flipped A tile in 64 VGPRs.
// ---------------------------------------------------------------------------
__global__ __launch_bounds__(32) void build_coupling_kernel(
    const float* __restrict__ emb, float* __restrict__ coup)
{
    const int id  = blockIdx.x;
    const int tjg = id % KN_GJ;
    const int ti  = (id / KN_GJ) % KN_NT;
    const int b   = id / (KN_GJ * KN_NT);

    const int lane = threadIdx.x;       // 0..31
    const int half = lane >> 4;         // 0 or 1
    const int l16  = lane & 15;
    const int koff = half * 2;          // K offset per ISA A-layout (16x4 f32)

    // ----- load + sign-flip the A tile (16 rows x 128 K) into registers -----
    const int nRow = ti * 16 + l16;
    const float* aBase = emb + ((size_t)b * KN_N + nRow) * KN_D + koff;

    v2f aReg[32];
#pragma unroll
    for (int kk = 0; kk < 32; ++kk) {
        v2f v = *(const v2f*)(aBase + 4 * kk);
        // column index of v.x is 4*kk + koff; only column 0 keeps its sign
        aReg[kk].x = (4 * kk + koff == 0) ? v.x : -v.x;
        aReg[kk].y = -v.y;              // column >= 1 always
    }

    // ----- sweep 16 column tiles -----
    for (int tj = tjg * KN_TJ; tj < tjg * KN_TJ + KN_TJ; ++tj) {
        const int mCol = tj * 16 + l16;
        const float* bBase = emb + ((size_t)b * KN_N + mCol) * KN_D + koff;

        v8f c = {0.f, 0.f, 0.f, 0.f, 0.f, 0.f, 0.f, 0.f};
#pragma unroll
        for (int kk = 0; kk < 32; ++kk) {
            v2f bv = *(const v2f*)(bBase + 4 * kk);
            c = __builtin_amdgcn_wmma_f32_16x16x4_f32(
                    /*neg_a=*/false, aReg[kk],
                    /*neg_b=*/false, bv,
                    /*c_mod=*/(short)0, c,
                    /*reuse_a=*/false, /*reuse_b=*/false);
        }

        // ----- epilogue: arg -> acosh -> exp -> clamp, then store -----
        // C/D layout: VGPR r holds M = r + 8*half, N = lane%16
#pragma unroll
        for (int r = 0; r < 8; ++r) {
            const int row = ti * 16 + r + half * 8;
            float arg = fmaxf(c[r], 1.0000001f);
            float dL  = acoshf(arg);
            float k   = __expf(-dL * INV_TAU);
            k = fminf(fmaxf(k, 0.0f), 1.0f);
            coup[((size_t)b * KN_N + row) * KN_N + mCol] = k;
        }
    }
}

// ---------------------------------------------------------------------------
// Kernel 2: one Kuramoto step (bandwidth-bound matvec from L2).
// Block = 256 threads (8 waves). Block caches sin/cos(theta) for its batch
// in LDS, each wave reduces one row at a time (4 rows per wave).
// ---------------------------------------------------------------------------
__global__ __launch_bounds__(256) void kuramoto_step_kernel(
    const float* __restrict__ thetaIn, const float* __restrict__ coup,
    const float* __restrict__ omega, float* __restrict__ thetaOut)
{
    __shared__ float sinL[KN_N];
    __shared__ float cosL[KN_N];

    const int b       = blockIdx.y;
    const int rowBase = blockIdx.x * 32;
    const float* th   = thetaIn + (size_t)b * KN_N;

    for (int m = threadIdx.x; m < KN_N; m += 256) {
        float s, c;
        sincosf(th[m], &s, &c);
        sinL[m] = s;
        cosL[m] = c;
    }
    __syncthreads();

    const int wave = threadIdx.x >> 5;
    const int lane = threadIdx.x & 31;
    const float4* sin4 = (const float4*)sinL;
    const float4* cos4 = (const float4*)cosL;

#pragma unroll
    for (int it = 0; it < 4; ++it) {
        const int n = rowBase + wave + 8 * it;
        const float* crow = coup + ((size_t)b * KN_N + n) * KN_N;

        float accC = 0.0f, accS = 0.0f;
        for (int m = lane * 4; m < KN_N; m += 128) {
            const float4 cv = *(const float4*)(crow + m);
            const float4 cs = cos4[m >> 2];
            const float4 sn = sin4[m >> 2];
            accC = fmaf(cv.x, cs.x, accC);
            accC = fmaf(cv.y, cs.y, accC);
            accC = fmaf(cv.z, cs.z, accC);
            accC = fmaf(cv.w, cs.w, accC);
            accS = fmaf(cv.x, sn.x, accS);
            accS = fmaf(cv.y, sn.y, accS);
            accS = fmaf(cv.z, sn.z, accS);
            accS = fmaf(cv.w, sn.w, accS);
        }

        // wave32 butterfly reduction
#pragma unroll
        for (int off = 16; off > 0; off >>= 1) {
            accC += __shfl_xor(accC, off, 32);
            accS += __shfl_xor(accS, off, 32);
        }

        if (lane == 0) {
            const float tn = th[n];
            const float coupling_sum = sinL[n] * accC - cosL[n] * accS;
            const float dtheta = omega[n] + K_OVER_N * coupling_sum;
            thetaOut[(size_t)b * KN_N + n] = fmodf(tn + DT_F * dtheta, TWO_PI_F);
        }
    }
}

// ---------------------------------------------------------------------------
// Launch: build coupling once, then 16 ping-pong step kernels.
// Workspace layout: [coupling: 8*2048*2048 f32][thetaA: 8*2048][thetaB: 8*2048]
// (needs ~128.1 MB of scratch)
// ---------------------------------------------------------------------------
extern "C" void kernel_launch(void* const* d_in, const int* in_sizes, int n_in,
                              void* d_out, int out_size, void* d_ws, size_t ws_size,
                              hipStream_t stream) {
    const float* initial = (const float*)d_in[0];   // (B, N)
    const float* emb     = (const float*)d_in[1];   // (B, N, D)
    const float* omega   = (const float*)d_in[2];   // (N,)
    float* out = (float*)d_out;                     // (B, N)

    float* coupling = (float*)d_ws;
    float* thetaA   = coupling + (size_t)KN_B * KN_N * KN_N;
    float* thetaB   = thetaA + (size_t)KN_B * KN_N;

    // Phase 1: coupling matrix via fp32 WMMA
    build_coupling_kernel<<<KN_B * KN_NT * KN_GJ, 32, 0, stream>>>(emb, coupling);

    // Phase 2: 16 steps, ping-pong, last writes d_out
    const float* cur = initial;
    for (int s = 0; s < 16; ++s) {
        float* dst = (s == 15) ? out : ((s & 1) ? thetaB : thetaA);
        kuramoto_step_kernel<<<dim3(KN_N / 32, KN_B), 256, 0, stream>>>(
            cur, coupling, omega, dst);
        cur = dst;
    }
}